// S4DLayer_87247965651560
// MI455X (gfx1250) — compile-verified
//
#include <hip/hip_runtime.h>
#include <cstdint>

// Problem geometry (matches reference: x [16, 4096, 512])
#define BB 16
#define LL 4096
#define DD 512
#define CHUNK 128                 // timesteps per chunk (pass1/pass3 unit)
#define NCHUNK (LL / CHUNK)       // 32
#define DTILE 256                 // channels per workgroup
#define NDT (DD / DTILE)          // 2
#define SROWS 16                  // timesteps staged per LDS buffer
#define NSTAGE (CHUNK / SROWS)    // 8
#define TPB 256
#define TILEF (SROWS * DTILE)     // floats per LDS buffer (4096 = 16KB)
#define APS ((SROWS * DTILE) / (4 * TPB)) // async b128 issues per thread per stage = 4

// ---- CDNA5 async global->LDS path (builtin if present, inline asm fallback) ----
#if defined(__HIP_DEVICE_COMPILE__) && defined(__has_builtin)
#if __has_builtin(__builtin_amdgcn_global_load_async_to_lds_b128)
#define HAVE_ASYNC_B 1
#endif
#if __has_builtin(__builtin_amdgcn_s_wait_asynccnt)
#define HAVE_WAIT_B 1
#endif
#endif

typedef int v4i __attribute__((ext_vector_type(4)));
typedef __attribute__((address_space(1))) v4i gv4i;   // global <4 x i32>
typedef __attribute__((address_space(3))) v4i lv4i;   // LDS    <4 x i32>

__device__ __forceinline__ void async_ld16(const float* g, unsigned lds_off) {
#if defined(HAVE_ASYNC_B)
  __builtin_amdgcn_global_load_async_to_lds_b128(
      (gv4i*)(uintptr_t)g, (lv4i*)(uintptr_t)lds_off, 0, 0);
#else
  asm volatile("global_load_async_to_lds_b128 %0, %1, off"
               :: "v"(lds_off), "v"(g) : "memory");
#endif
}

__device__ __forceinline__ void wait_async_leN() {  // <= APS outstanding
#if defined(HAVE_WAIT_B)
  __builtin_amdgcn_s_wait_asynccnt(APS);
#else
  asm volatile("s_wait_asynccnt %0" :: "n"(APS));
#endif
}
__device__ __forceinline__ void wait_async_0() {
#if defined(HAVE_WAIT_B)
  __builtin_amdgcn_s_wait_asynccnt(0);
#else
  asm volatile("s_wait_asynccnt 0");
#endif
}

// Issue one 16-row x-tile into LDS buffer at byte base lbase.
__device__ __forceinline__ void stage_issue(const float* __restrict__ gbase,
                                            unsigned lbase, int tid) {
#pragma unroll
  for (int k = 0; k < APS; ++k) {
    int c   = tid + k * TPB;        // 0..1023 b128 chunks
    int row = c >> 6;               // 64 chunks per row (256 floats)
    int col = (c & 63) << 2;        // float column
    async_ld16(gbase + (size_t)row * DD + col,
               lbase + (unsigned)(((row << 8) + col) << 2));
  }
}

// ---------------- Pass 1: per-chunk local end state S (h0 = 0) ----------------
__global__ void __launch_bounds__(TPB) s4d_pass1(
    const float* __restrict__ x, const float* __restrict__ Av,
    const float* __restrict__ Bv, float* __restrict__ Sws) {
  __shared__ float tile[2 * TILEF];
  const int tid   = threadIdx.x;
  const int bid   = blockIdx.x;
  const int dt    = bid & (NDT - 1);
  const int chunk = (bid >> 1) & (NCHUNK - 1);
  const int b     = bid >> 6;                 // NDT*NCHUNK == 64
  const int dbase = dt * DTILE;
  const int d     = dbase + tid;

  const float dec = expf(Av[d]);
  const float Bd  = Bv[d];
  const int   t0  = chunk * CHUNK;
  const float* gx = x + ((size_t)b * LL + t0) * DD + dbase;
  const unsigned lds0 = (unsigned)(uintptr_t)&tile[0];

  stage_issue(gx, lds0, tid);                 // prime stage 0 -> buffer 0

  float S = 0.f;
  for (int s = 0; s < NSTAGE; ++s) {
    if (s + 1 < NSTAGE) {
      stage_issue(gx + (size_t)(s + 1) * SROWS * DD,
                  lds0 + (unsigned)(((s + 1) & 1) * TILEF * 4), tid);
      wait_async_leN();                       // stage s complete (own wave)
    } else {
      wait_async_0();
    }
    __syncthreads();                          // stage s visible from all waves
    const int off = (s & 1) * TILEF;
#pragma unroll
    for (int r = 0; r < SROWS; ++r) {
      float xv = tile[off + (r << 8) + tid];
      S = fmaf(S, dec, Bd * xv);
    }
    __syncthreads();                          // free buffer for stage s+2
  }
  Sws[((size_t)b * NCHUNK + chunk) * DD + d] = S;
}

// --------- Pass 2: scan over chunks -> entry states H_k and h_last ----------
__global__ void __launch_bounds__(TPB) s4d_pass2(
    const float* __restrict__ Av, const float* __restrict__ Sws,
    float* __restrict__ Hws, float* __restrict__ hlast) {
  const int idx = blockIdx.x * TPB + threadIdx.x;   // 0..BB*DD-1
  const int b = idx / DD;
  const int d = idx - b * DD;
  const float decT = expf(Av[d] * (float)CHUNK);    // e^{A*CHUNK} = decay^CHUNK
  float h = 0.f;
  for (int k = 0; k < NCHUNK; ++k) {
    const size_t o = ((size_t)b * NCHUNK + k) * DD + d;
    Hws[o] = h;                                     // entry state of chunk k
    h = fmaf(h, decT, Sws[o]);
  }
  hlast[(size_t)b * DD + d] = h;
}

// ------- Pass 3: replay each chunk from correct entry state, emit y ---------
__global__ void __launch_bounds__(TPB) s4d_pass3(
    const float* __restrict__ x, const float* __restrict__ Av,
    const float* __restrict__ Bv, const float* __restrict__ Cv,
    const float* __restrict__ Dv, const float* __restrict__ Hws,
    float* __restrict__ y) {
  __shared__ float tile[2 * TILEF];
  const int tid   = threadIdx.x;
  const int bid   = blockIdx.x;
  const int dt    = bid & (NDT - 1);
  const int chunk = (bid >> 1) & (NCHUNK - 1);
  const int b     = bid >> 6;
  const int dbase = dt * DTILE;
  const int d     = dbase + tid;

  const float dec = expf(Av[d]);
  const float Bd  = Bv[d];
  const float Cd  = Cv[d];
  const float Dd  = Dv[d];
  const int   t0  = chunk * CHUNK;
  const float* gx = x + ((size_t)b * LL + t0) * DD + dbase;
  float*       gy = y + ((size_t)b * LL + t0) * DD + d;  // this thread's column
  const unsigned lds0 = (unsigned)(uintptr_t)&tile[0];

  float h = Hws[((size_t)b * NCHUNK + chunk) * DD + d];

  stage_issue(gx, lds0, tid);
  for (int s = 0; s < NSTAGE; ++s) {
    if (s + 1 < NSTAGE) {
      stage_issue(gx + (size_t)(s + 1) * SROWS * DD,
                  lds0 + (unsigned)(((s + 1) & 1) * TILEF * 4), tid);
      wait_async_leN();
    } else {
      wait_async_0();
    }
    __syncthreads();
    const int off = (s & 1) * TILEF;
#pragma unroll
    for (int r = 0; r < SROWS; ++r) {
      float xv = tile[off + (r << 8) + tid];
      h = fmaf(h, dec, Bd * xv);
      gy[(size_t)(s * SROWS + r) * DD] = fmaf(Cd, h, Dd * xv);
    }
    __syncthreads();
  }
}

extern "C" void kernel_launch(void* const* d_in, const int* in_sizes, int n_in,
                              void* d_out, int out_size, void* d_ws, size_t ws_size,
                              hipStream_t stream) {
  (void)in_sizes; (void)n_in; (void)out_size; (void)ws_size;
  const float* x  = (const float*)d_in[0];
  const float* Av = (const float*)d_in[1];
  const float* Bv = (const float*)d_in[2];
  const float* Cv = (const float*)d_in[3];
  const float* Dv = (const float*)d_in[4];

  float* y     = (float*)d_out;
  float* hlast = y + (size_t)BB * LL * DD;           // outputs concatenated: y then h_last

  float* Sws = (float*)d_ws;                          // [BB, NCHUNK, DD]  (1 MB)
  float* Hws = Sws + (size_t)BB * NCHUNK * DD;        // [BB, NCHUNK, DD]  (1 MB)

  const int grid13 = BB * NCHUNK * NDT;               // 1024 workgroups
  s4d_pass1<<<grid13, TPB, 0, stream>>>(x, Av, Bv, Sws);
  s4d_pass2<<<(BB * DD) / TPB, TPB, 0, stream>>>(Av, Sws, Hws, hlast);
  s4d_pass3<<<grid13, TPB, 0, stream>>>(x, Av, Bv, Cv, Dv, Hws, y);
}